// DeformableAttention_20435454395189
// MI455X (gfx1250) — compile-verified
//
#include <hip/hip_runtime.h>
#include <hip/hip_bf16.h>

// Shapes from reference
#define BATCH 4
#define CCH   384
#define HH    56
#define WW    56
#define NHEAD 12
#define HCH   32
#define HK    28
#define WK    28
#define NPOS  (HK * WK)      // 784  (key/off spatial)
#define MFULL (HH * WW)      // 3136 (query spatial)
#define SCALE 0.1767766952966369f  // 32^-0.5

typedef __attribute__((ext_vector_type(2))) float v2f;
typedef __attribute__((ext_vector_type(8))) float v8f;

__device__ __forceinline__ float wave_sum(float v) {
    #pragma unroll
    for (int off = 16; off > 0; off >>= 1)
        v += __shfl_xor(v, off, 32);
    return v;
}

// ---------------------------------------------------------------------------
// K1: fused depthwise 3x3, stride 1 -> value, query   (one thread per pixel)
// ---------------------------------------------------------------------------
__global__ void dw_s1_kernel(const float* __restrict__ x,
                             const float* __restrict__ w_v, const float* __restrict__ b_v,
                             const float* __restrict__ w_q, const float* __restrict__ b_q,
                             float* __restrict__ value, float* __restrict__ query) {
    int tid = blockIdx.x * blockDim.x + threadIdx.x;
    if (tid >= BATCH * CCH * HH * WW) return;
    int xx = tid % WW;
    int yy = (tid / WW) % HH;
    int c  = (tid / (HH * WW)) % CCH;
    int bc = tid / (HH * WW);                  // b*CCH + c
    const float* xp = x + (size_t)bc * (HH * WW);
    const float* wv = w_v + c * 9;
    const float* wq = w_q + c * 9;
    float av = 0.f, aq = 0.f;
    #pragma unroll
    for (int dy = 0; dy < 3; ++dy) {
        int iy = yy + dy - 1;
        if (iy < 0 || iy >= HH) continue;
        #pragma unroll
        for (int dx = 0; dx < 3; ++dx) {
            int ix = xx + dx - 1;
            if (ix < 0 || ix >= WW) continue;
            float xv = xp[iy * WW + ix];
            av += xv * wv[dy * 3 + dx];
            aq += xv * wq[dy * 3 + dx];
        }
    }
    value[tid] = av + b_v[c];
    query[tid] = aq + b_q[c];
}

// ---------------------------------------------------------------------------
// K2: fused depthwise 3x3, stride 2 -> key, off
// ---------------------------------------------------------------------------
__global__ void dw_s2_kernel(const float* __restrict__ x,
                             const float* __restrict__ w_k, const float* __restrict__ b_k,
                             const float* __restrict__ w_o, const float* __restrict__ b_o,
                             float* __restrict__ key, float* __restrict__ off) {
    int tid = blockIdx.x * blockDim.x + threadIdx.x;
    if (tid >= BATCH * CCH * NPOS) return;
    int kx = tid % WK;
    int ky = (tid / WK) % HK;
    int c  = (tid / NPOS) % CCH;
    int bc = tid / NPOS;                       // b*CCH + c
    const float* xp = x + (size_t)bc * (HH * WW);
    const float* wk = w_k + c * 9;
    const float* wo = w_o + c * 9;
    float ak = 0.f, ao = 0.f;
    #pragma unroll
    for (int dy = 0; dy < 3; ++dy) {
        int iy = 2 * ky + dy - 1;
        if (iy < 0 || iy >= HH) continue;
        #pragma unroll
        for (int dx = 0; dx < 3; ++dx) {
            int ix = 2 * kx + dx - 1;
            if (ix < 0 || ix >= WW) continue;
            float xv = xp[iy * WW + ix];
            ak += xv * wk[dy * 3 + dx];
            ao += xv * wo[dy * 3 + dx];
        }
    }
    key[tid] = ak + b_k[c];
    off[tid] = ao + b_o[c];
}

// ---------------------------------------------------------------------------
// K3: per-position LayerNorm(C) + exact GELU + 2xC offset projection + tanh
//     grid + bilinear sample of `value` -> v_s.  One wave32 per (b,ky,kx).
// ---------------------------------------------------------------------------
__global__ void offset_sample_kernel(const float* __restrict__ off,
                                     const float* __restrict__ ln_g,
                                     const float* __restrict__ ln_b,
                                     const float* __restrict__ w_off2,
                                     const float* __restrict__ value,
                                     float* __restrict__ v_s) {
    int wid  = blockIdx.x * (blockDim.x >> 5) + (threadIdx.x >> 5);  // 0..3135 exact
    int lane = threadIdx.x & 31;
    int b = wid / NPOS;
    int s = wid % NPOS;
    int ky = s / WK, kx = s % WK;

    float ov[12];
    float sum = 0.f, sq = 0.f;
    #pragma unroll
    for (int i = 0; i < 12; ++i) {
        int c = lane + 32 * i;
        float v = off[(size_t)(b * CCH + c) * NPOS + s];
        ov[i] = v; sum += v; sq += v * v;
    }
    sum = wave_sum(sum);
    sq  = wave_sum(sq);
    float mean = sum * (1.f / CCH);
    float var  = sq * (1.f / CCH) - mean * mean;
    float rstd = rsqrtf(var + 1e-5f);

    float d0 = 0.f, d1 = 0.f;
    #pragma unroll
    for (int i = 0; i < 12; ++i) {
        int c = lane + 32 * i;
        float g = (ov[i] - mean) * rstd * ln_g[c] + ln_b[c];
        g = 0.5f * g * (1.f + erff(g * 0.70710678118654752f));   // exact GELU
        d0 += g * w_off2[c];
        d1 += g * w_off2[CCH + c];
    }
    d0 = wave_sum(d0);
    d1 = wave_sum(d1);

    float ry = ((float)ky + 0.5f) / (float)HK * 2.f - 1.f;
    float rx = ((float)kx + 0.5f) / (float)WK * 2.f - 1.f;
    float py = tanhf(d0 + ry);          // pos[...,0] (y)
    float px = tanhf(d1 + rx);          // pos[...,1] (x); grid = pos[...,::-1]
    float ix = (px + 1.f) * 0.5f * (float)(WW - 1);
    float iy = (py + 1.f) * 0.5f * (float)(HH - 1);
    float x0 = floorf(ix), y0 = floorf(iy);
    float wx = ix - x0,    wy = iy - y0;
    int x0i = min(max((int)x0, 0), WW - 1);
    int x1i = min(max((int)x0 + 1, 0), WW - 1);
    int y0i = min(max((int)y0, 0), HH - 1);
    int y1i = min(max((int)y0 + 1, 0), HH - 1);
    float w00 = (1.f - wx) * (1.f - wy), w01 = wx * (1.f - wy);
    float w10 = (1.f - wx) * wy,         w11 = wx * wy;

    #pragma unroll
    for (int i = 0; i < 12; ++i) {
        int c = lane + 32 * i;
        const float* vp = value + (size_t)(b * CCH + c) * (HH * WW);
        float r = vp[y0i * WW + x0i] * w00 + vp[y0i * WW + x1i] * w01 +
                  vp[y1i * WW + x0i] * w10 + vp[y1i * WW + x1i] * w11;
        v_s[(size_t)(b * CCH + c) * NPOS + s] = r;
    }
}

// ---------------------------------------------------------------------------
// K4: KV[head] = SCALE * key(32x784) @ v_s(32x784)^T   (48 heads, 32x32 each)
//     One 16x16 WMMA tile per wave, 4 waves per head.  K-loop: 784 in steps 4.
// ---------------------------------------------------------------------------
__global__ void kv_kernel(const float* __restrict__ key,
                          const float* __restrict__ v_s,
                          float* __restrict__ KV) {
    int head = blockIdx.x;                    // 0..47
    int w    = threadIdx.x >> 5;              // 0..3
    int lane = threadIdx.x & 31;
    int half = lane >> 4;                     // 0|1
    int ln   = lane & 15;
    int ti = w >> 1, tj = w & 1;
    int b = head / NHEAD, h = head % NHEAD;

    const float* kbase = key + (size_t)(b * CCH + h * HCH + ti * 16 + ln) * NPOS;
    const float* vbase = v_s + (size_t)(b * CCH + h * HCH + tj * 16 + ln) * NPOS;

    v8f acc = {};
    for (int n0 = 0; n0 < NPOS; n0 += 4) {
        // A[m][k]: m = key channel (row ln), k = spatial. lanes 16-31 carry K=2,3.
        v2f a  = *(const v2f*)(kbase + n0 + 2 * half);
        // B[k][n]: n = v channel (col ln), k = spatial (contiguous per lane).
        v2f bb = *(const v2f*)(vbase + n0 + 2 * half);
        acc = __builtin_amdgcn_wmma_f32_16x16x4_f32(false, a, false, bb,
                                                    (short)0, acc, false, false);
    }
    float* out = KV + (size_t)head * HCH * HCH;
    #pragma unroll
    for (int r = 0; r < 8; ++r) {
        int ck = ti * 16 + r + 8 * half;   // row from A (key channel)
        int cv = tj * 16 + ln;             // col from B (v channel)
        out[ck * HCH + cv] = acc[r] * SCALE;
    }
}

// ---------------------------------------------------------------------------
// K5: W2[b][:, h*32+c'] = w_out[:, h*32: h*32+32] @ KV[b,h]^T   (K = 32)
//     2304 tiles of 16x16, one per wave.
// ---------------------------------------------------------------------------
__global__ void w2_kernel(const float* __restrict__ w_out,
                          const float* __restrict__ KV,
                          float* __restrict__ W2) {
    int wid  = blockIdx.x * 4 + (threadIdx.x >> 5);  // 0..2303 exact
    int lane = threadIdx.x & 31;
    int half = lane >> 4, ln = lane & 15;
    int bh = wid / 48;                               // batch*NHEAD
    int r  = wid % 48;
    int mt = r >> 1, nt = r & 1;
    int b = bh / NHEAD, h = bh % NHEAD;

    const float* A  = w_out + (size_t)(mt * 16 + ln) * CCH + h * HCH;      // A[m][k]
    const float* Bm = KV + (size_t)bh * HCH * HCH + (nt * 16 + ln) * HCH;  // B[k][n]=KV[n][k]

    v8f acc = {};
    #pragma unroll
    for (int k0 = 0; k0 < HCH; k0 += 4) {
        v2f a  = *(const v2f*)(A  + k0 + 2 * half);
        v2f bb = *(const v2f*)(Bm + k0 + 2 * half);
        acc = __builtin_amdgcn_wmma_f32_16x16x4_f32(false, a, false, bb,
                                                    (short)0, acc, false, false);
    }
    float* out = W2 + (size_t)b * CCH * CCH;
    #pragma unroll
    for (int rr = 0; rr < 8; ++rr) {
        int m = mt * 16 + rr + 8 * half;
        int n = h * HCH + nt * 16 + ln;
        out[m * CCH + n] = acc[rr];
    }
}

// ---------------------------------------------------------------------------
// K6: y[b] = W2[b] (384x384) @ Q[b] (384x3136).  Each wave owns a 16x64 tile
//     (4 accumulators share one A fragment).  Exact tiling: 24 x 49 x 4 batches.
// ---------------------------------------------------------------------------
__global__ void final_gemm_kernel(const float* __restrict__ W2,
                                  const float* __restrict__ Q,
                                  float* __restrict__ y) {
    int wid  = blockIdx.x * 4 + (threadIdx.x >> 5);  // 0..4703 exact
    int lane = threadIdx.x & 31;
    int half = lane >> 4, ln = lane & 15;
    int b  = wid / (24 * 49);
    int r  = wid % (24 * 49);
    int mt = r / 49, nt = r % 49;
    int m0 = mt * 16, n0 = nt * 64;

    const float* A  = W2 + (size_t)(b * CCH + m0 + ln) * CCH;
    const float* Bq = Q + (size_t)b * CCH * MFULL;

    v8f acc0 = {}, acc1 = {}, acc2 = {}, acc3 = {};
    for (int k0 = 0; k0 < CCH; k0 += 4) {
        v2f a = *(const v2f*)(A + k0 + 2 * half);
        const float* brow0 = Bq + (size_t)(k0 + 2 * half) * MFULL + n0 + ln;
        const float* brow1 = brow0 + MFULL;
        v2f b0 = { brow0[0],  brow1[0]  };
        v2f b1 = { brow0[16], brow1[16] };
        v2f b2 = { brow0[32], brow1[32] };
        v2f b3 = { brow0[48], brow1[48] };
        acc0 = __builtin_amdgcn_wmma_f32_16x16x4_f32(false, a, false, b0, (short)0, acc0, false, false);
        acc1 = __builtin_amdgcn_wmma_f32_16x16x4_f32(false, a, false, b1, (short)0, acc1, false, false);
        acc2 = __builtin_amdgcn_wmma_f32_16x16x4_f32(false, a, false, b2, (short)0, acc2, false, false);
        acc3 = __builtin_amdgcn_wmma_f32_16x16x4_f32(false, a, false, b3, (short)0, acc3, false, false);
    }
    float* yb = y + (size_t)(b * CCH + m0) * MFULL + n0;
    #pragma unroll
    for (int rr = 0; rr < 8; ++rr) {
        int mrow = rr + 8 * half;
        float* yr = yb + (size_t)mrow * MFULL + ln;
        yr[0]  = acc0[rr];
        yr[16] = acc1[rr];
        yr[32] = acc2[rr];
        yr[48] = acc3[rr];
    }
}

// ---------------------------------------------------------------------------
extern "C" void kernel_launch(void* const* d_in, const int* in_sizes, int n_in,
                              void* d_out, int out_size, void* d_ws, size_t ws_size,
                              hipStream_t stream) {
    const float* x      = (const float*)d_in[0];
    const float* w_v    = (const float*)d_in[1];
    const float* b_v    = (const float*)d_in[2];
    const float* w_q    = (const float*)d_in[3];
    const float* b_q    = (const float*)d_in[4];
    const float* w_k    = (const float*)d_in[5];
    const float* b_k    = (const float*)d_in[6];
    const float* w_off1 = (const float*)d_in[7];
    const float* b_off1 = (const float*)d_in[8];
    const float* ln_g   = (const float*)d_in[9];
    const float* ln_b   = (const float*)d_in[10];
    const float* w_off2 = (const float*)d_in[11];
    const float* w_out  = (const float*)d_in[12];

    // Workspace layout (floats). Total = 13,885,440 floats ~= 55.5 MB.
    float* ws    = (float*)d_ws;
    float* value = ws;                       // 4,816,896
    float* query = value + 4816896;          // 4,816,896
    float* key   = query + 4816896;          // 1,204,224
    float* off   = key   + 1204224;          // 1,204,224
    float* v_s   = off   + 1204224;          // 1,204,224
    float* KV    = v_s   + 1204224;          //    49,152
    float* W2    = KV    + 49152;            //   589,824

    dw_s1_kernel<<<(BATCH * CCH * HH * WW) / 256, 256, 0, stream>>>(
        x, w_v, b_v, w_q, b_q, value, query);
    dw_s2_kernel<<<(BATCH * CCH * NPOS) / 256, 256, 0, stream>>>(
        x, w_k, b_k, w_off1, b_off1, key, off);
    offset_sample_kernel<<<(BATCH * NPOS) / 4, 128, 0, stream>>>(
        off, ln_g, ln_b, w_off2, value, v_s);
    kv_kernel<<<BATCH * NHEAD, 128, 0, stream>>>(key, v_s, KV);
    w2_kernel<<<576, 128, 0, stream>>>(w_out, KV, W2);
    final_gemm_kernel<<<1176, 128, 0, stream>>>(W2, query, (float*)d_out);
}